// AuxCMP_61907658604772
// MI455X (gfx1250) — compile-verified
//
#include <hip/hip_runtime.h>
#include <stdint.h>

// Problem constants (fixed by the reference: B=64, T=12, C=64, SIDE=32).
#define KB   64
#define KT   12
#define KC   64
#define KS2  1024                 // SIDE*SIDE
#define KROWS (KB * KC)           // 4096 (b,c) rows in the last time slice
#define ROWS_PER_BLOCK 32

typedef int v4i __attribute__((ext_vector_type(4)));

// Element base of row r (r = b*KC + c) inside the LAST time slice of a
// (B,T,C,S2) tensor: ((b*T + (T-1))*C + c) * S2.  Max < 2^26, fits in int.
__device__ __forceinline__ int lastSliceRowBase(int r) {
    int b = r >> 6;            // r / KC
    int c = r & (KC - 1);      // r % KC
    return ((b * KT + (KT - 1)) * KC + c) * KS2;
}

// ---- CDNA5 async global->LDS copy (ASYNCcnt path) -------------------------
// Builtin signature (from hipcc diagnostic): param0 = int4-vector pointer in
// the device/global address space, so cast to the exact expected types.
__device__ __forceinline__ void asyncCopy16B(const float* gsrc, float* ldst) {
#if defined(__has_builtin) && __has_builtin(__builtin_amdgcn_global_load_async_to_lds_b128)
    __builtin_amdgcn_global_load_async_to_lds_b128(
        (__attribute__((address_space(1))) v4i*)gsrc,
        (__attribute__((address_space(3))) v4i*)ldst,
        /*imm offset=*/0, /*cpol=*/0);
#else
    *(float4*)ldst = *(const float4*)gsrc;   // fallback: synchronous copy
#endif
}

__device__ __forceinline__ void waitAsync() {
#if defined(__has_builtin) && __has_builtin(__builtin_amdgcn_s_wait_asynccnt)
    __builtin_amdgcn_s_wait_asynccnt(0);
#else
    asm volatile("s_wait_asynccnt 0" ::: "memory");
#endif
}

// ---- Kernel 1: zero the per-cell OR accumulator (1024 ints in d_ws) -------
__global__ void auxcmp_init(int* __restrict__ ws) {
    int4 z = {0, 0, 0, 0};
    ((int4*)ws)[threadIdx.x] = z;            // 256 threads * 16B = 4KB
}

// ---- Kernel 2: cell_empty reduction: OR of mask over all 4096 rows --------
// Grid: KROWS/ROWS_PER_BLOCK = 128 blocks, 256 threads.
// Thread t owns cells [4t, 4t+3]; coalesced int4 loads (one B128 per thread
// per row covers the whole 4KB mask row across the block).
__global__ void auxcmp_mask_reduce(const int* __restrict__ m,
                                   int* __restrict__ ws) {
    const int tid = threadIdx.x;
    const int row0 = blockIdx.x * ROWS_PER_BLOCK;

    int4 acc = {0, 0, 0, 0};
#pragma unroll 4
    for (int i = 0; i < ROWS_PER_BLOCK; ++i) {
        const int4 v = *((const int4*)(m + lastSliceRowBase(row0 + i)) + tid);
        acc.x |= v.x; acc.y |= v.y; acc.z |= v.z; acc.w |= v.w;
    }
    const int s = tid * 4;
    if (acc.x) atomicOr(ws + s + 0, 1);
    if (acc.y) atomicOr(ws + s + 1, 1);
    if (acc.z) atomicOr(ws + s + 2, 1);
    if (acc.w) atomicOr(ws + s + 3, 1);
}

// ---- Kernel 3: out = data + (cell_empty ? data[poi] : 0) ------------------
// One block per (b,c) row. Async-DMA the 4KB row into LDS, then the gather
// data[b,c,poi[s]] becomes an LDS read (poi[s] < 1024 indexes THIS row).
__global__ void auxcmp_gather_add(const float* __restrict__ d,
                                  const int* __restrict__ flags,     // 0 == empty
                                  const long long* __restrict__ poi,
                                  float* __restrict__ out) {
    __shared__ float row[KS2];               // 4 KB

    const int tid = threadIdx.x;             // 256 threads (8 wave32s)
    const int r   = blockIdx.x;              // row = b*KC + c
    const float* src = d + lastSliceRowBase(r);

    // 256 lanes x 16B = the whole 4KB row, via the async-to-LDS path.
    asyncCopy16B(src + tid * 4, &row[tid * 4]);
    waitAsync();
    __syncthreads();

    const int s0 = tid * 4;
    const int4 f = *(const int4*)(flags + s0);
    const longlong2 p01 = *(const longlong2*)(poi + s0);
    const longlong2 p23 = *(const longlong2*)(poi + s0 + 2);
    const float4 v = *(const float4*)(&row[s0]);

    float4 o;
    o.x = v.x + (f.x == 0 ? row[(int)p01.x] : 0.0f);
    o.y = v.y + (f.y == 0 ? row[(int)p01.y] : 0.0f);
    o.z = v.z + (f.z == 0 ? row[(int)p23.x] : 0.0f);
    o.w = v.w + (f.w == 0 ? row[(int)p23.y] : 0.0f);

    *(float4*)(out + (size_t)r * KS2 + s0) = o;
}

extern "C" void kernel_launch(void* const* d_in, const int* in_sizes, int n_in,
                              void* d_out, int out_size, void* d_ws, size_t ws_size,
                              hipStream_t stream) {
    (void)in_sizes; (void)n_in; (void)out_size; (void)ws_size;

    const float*     d_data = (const float*)d_in[0];      // (B,T,C,S2) f32
    const int*       d_mask = (const int*)d_in[1];        // (B,T,C,S2) i32
    const long long* d_poi  = (const long long*)d_in[2];  // (S2,) i64
    // d_in[3] = side (always 32) — unused.

    int*   ws  = (int*)d_ws;                              // 1024-int OR accumulator
    float* out = (float*)d_out;                           // (B,C,32,32) f32

    auxcmp_init<<<1, 256, 0, stream>>>(ws);
    auxcmp_mask_reduce<<<KROWS / ROWS_PER_BLOCK, 256, 0, stream>>>(d_mask, ws);
    auxcmp_gather_add<<<KROWS, 256, 0, stream>>>(d_data, ws, d_poi, out);
}